// NAT_15857019257408
// MI455X (gfx1250) — compile-verified
//
#include <hip/hip_runtime.h>
#include <hip/hip_bf16.h>

// ---------------------------------------------------------------------------
// Neighborhood attention (NAT) for MI455X / gfx1250, wave32.
// fp32 WMMA (V_WMMA_F32_16X16X4_F32) for all three GEMMs (exact fp32 ordering
// for the top-k). Operand feed: b64-vectorized per-lane fragment loads;
// weight panels staged transposed in LDS and shared by 8 waves per block.
// ---------------------------------------------------------------------------

typedef __attribute__((ext_vector_type(2))) float v2f;
typedef __attribute__((ext_vector_type(8))) float v8f;

#define BATCH    2
#define S_LEN    2048
#define D_MODEL  256
#define N_HEADS  8
#define HEAD_DIM 32
#define BH       (BATCH * N_HEADS)     // 16
#define N3       (3 * D_MODEL)         // 768
#define LS       16                    // neighbors
#define M_ROWS   (BATCH * S_LEN)       // 4096
#define WPAD     260                   // 256 + 4 pad: conflict-free ds_load_b64

__device__ __forceinline__ v8f wmma_f32(v2f a, v2f b, v8f c) {
  // 8-arg form: (neg_a, A, neg_b, B, c_mod, C, reuse_a, reuse_b)
  return __builtin_amdgcn_wmma_f32_16x16x4_f32(false, a, false, b, (short)0, c,
                                               false, false);
}

// ---------------------------------------------------------------------------
// Kernel 1: QKV projection. Block = 8 m-tiles x 1 n-tile; the K x 16 column
// panel of w is staged transposed in LDS once and shared by all 8 waves.
// Output scattered into per-head q/k/v [bh][s][hd] layout, bias fused.
// ---------------------------------------------------------------------------
__global__ void __launch_bounds__(256)
qkv_gemm_kernel(const float* __restrict__ x,      // [4096,256]
                const float* __restrict__ w,      // [256,768]
                const float* __restrict__ bias,   // [768]
                float* __restrict__ q, float* __restrict__ k,
                float* __restrict__ v) {
  __shared__ float wT[16][WPAD];       // wT[col][k] = w[k][n0+col]
  const int lane = threadIdx.x & 31;
  const int wave = threadIdx.x >> 5;
  const int nt   = blockIdx.x % (N3 / 16);
  const int mgrp = blockIdx.x / (N3 / 16);
  const int m0 = (mgrp * 8 + wave) * 16;
  const int n0 = nt * 16;
  const int row = lane & 15;           // M (for A) / N (for B,C)
  const int hi2 = (lane >> 4) * 2;     // 0: K+{0,1} | 2: K+{2,3}

  {  // cooperative, coalesced panel load (16 cols x 256 k)
    const int col = threadIdx.x & 15;
    const int kb  = threadIdx.x >> 4;  // 0..15
#pragma unroll
    for (int kk = 0; kk < 16; ++kk) {
      const int kidx = kb + kk * 16;
      wT[col][kidx] = w[kidx * N3 + n0 + col];
    }
  }
  __syncthreads();

  v8f acc = {};
  const float* ap = x + (m0 + row) * D_MODEL;
#pragma unroll 4
  for (int k0 = 0; k0 < D_MODEL; k0 += 4) {
    v2f a = *(const v2f*)(ap + k0 + hi2);             // global_load_b64
    v2f b = *(const v2f*)&wT[row][k0 + hi2];          // ds_load_b64
    acc = wmma_f32(a, b, acc);
  }

  const int n  = n0 + row;             // output column (per lane)
  const float bv = bias[n];
  const int which = n / D_MODEL;       // 0=q 1=k 2=v
  const int c  = n % D_MODEL;
  const int h  = c / HEAD_DIM;
  const int hd = c % HEAD_DIM;
  float* dst = (which == 0) ? q : (which == 1) ? k : v;
#pragma unroll
  for (int r = 0; r < 8; ++r) {
    const int m  = m0 + r + (hi2 << 2);  // C/D layout: M split at lane 16
    const int bb = m >> 11;              // / 2048
    const int s  = m & (S_LEN - 1);
    dst[(((bb * N_HEADS + h) * S_LEN) + s) * HEAD_DIM + hd] = acc[r] + bv;
  }
}

// ---------------------------------------------------------------------------
// Kernel 2: squared norms of Q rows.
// ---------------------------------------------------------------------------
__global__ void __launch_bounds__(256)
sqnorm_kernel(const float* __restrict__ q, float* __restrict__ sq) {
  const int i = blockIdx.x * blockDim.x + threadIdx.x;  // 0..BH*S-1
  const float* p = q + i * HEAD_DIM;
  float s = 0.f;
#pragma unroll
  for (int d = 0; d < HEAD_DIM; ++d) s += p[d] * p[d];
  sq[i] = s;
}

// ---------------------------------------------------------------------------
// Kernel 3: pairwise-distance top-16. One wave = 16 query rows of one (b,h).
// Gram tile via 8x WMMA f32 16x16x4 (K=32), distance = sq_i + sq_j - 2*g.
// Lanes 0..15 keep a sorted 16-entry (dist,idx) list in registers.
// ---------------------------------------------------------------------------
__global__ void __launch_bounds__(256)
topk_kernel(const float* __restrict__ q, const float* __restrict__ sq,
            int* __restrict__ idx_out) {
  __shared__ float lds[8][256];        // 16x16 Gram tile per wave
  const int lane = threadIdx.x & 31;
  const int wave = threadIdx.x >> 5;
  const int wtile = blockIdx.x * 8 + wave;   // 0..BH*128-1
  const int bh = wtile >> 7;
  const int s0 = (wtile & 127) * 16;
  const int row = lane & 15;
  const int hi2 = (lane >> 4) * 2;

  const float* qb  = q + bh * S_LEN * HEAD_DIM;
  const float* sqb = sq + bh * S_LEN;

  // A fragments for the 16 query rows (hd=32 -> 8 K-steps), kept resident.
  v2f afr[8];
  const float* arow = qb + (s0 + row) * HEAD_DIM;
#pragma unroll
  for (int kk = 0; kk < 8; ++kk)
    afr[kk] = *(const v2f*)(arow + kk * 4 + hi2);     // global_load_b64
  const float sq_row = sqb[s0 + row];

  float best_d[LS];
  int   best_i[LS];
#pragma unroll
  for (int t = 0; t < LS; ++t) { best_d[t] = 3.4e38f; best_i[t] = 0; }

  for (int j = 0; j < S_LEN / 16; ++j) {
    const int key0 = j * 16;
    const float* brow = qb + (key0 + row) * HEAD_DIM;
    if (j + 1 < S_LEN / 16)  // hint next key tile toward WGP$ (global_prefetch)
      __builtin_prefetch(brow + 16 * HEAD_DIM, 0, 1);
    v8f acc = {};
#pragma unroll
    for (int kk = 0; kk < 8; ++kk) {
      // B = (key tile)^T : B[k][n] = Q[key0+n][k] -> same load pattern as A.
      v2f bfr = *(const v2f*)(brow + kk * 4 + hi2);   // global_load_b64
      acc = wmma_f32(afr[kk], bfr, acc);
    }
#pragma unroll
    for (int r = 0; r < 8; ++r)
      lds[wave][(r + (hi2 << 2)) * 16 + row] = acc[r];
    __syncthreads();

    if (lane < 16) {
#pragma unroll
      for (int c2 = 0; c2 < 16; ++c2) {
        const int key = key0 + c2;
        const float g  = lds[wave][row * 16 + c2];
        float cd = sq_row + sqb[key] - 2.0f * g;   // sqb[key] lane-invariant
        if (cd < best_d[LS - 1]) {                 // guarded insertion
          int ci = key;
#pragma unroll
          for (int t = 0; t < LS; ++t) {
            if (cd < best_d[t]) {                  // strict <: lower idx wins tie
              float td = best_d[t]; int ti = best_i[t];
              best_d[t] = cd; best_i[t] = ci;
              cd = td; ci = ti;
            }
          }
        }
      }
    }
    __syncthreads();
  }

  if (lane < 16) {
    int* op = idx_out + (bh * S_LEN + s0 + row) * LS;
#pragma unroll
    for (int t = 0; t < LS; ++t) op[t] = best_i[t];
  }
}

// ---------------------------------------------------------------------------
// Kernel 4: gathered 16-neighbor attention. One wave per query.
// Lanes 0..15 own one neighbor each (logit + softmax via width-16 shfl_xor,
// duplicated into lanes 16..31); all 32 lanes own one output dim.
// ---------------------------------------------------------------------------
__global__ void __launch_bounds__(256)
local_attn_kernel(const float* __restrict__ q, const float* __restrict__ k,
                  const float* __restrict__ v, const int* __restrict__ idx,
                  float* __restrict__ out) {
  const int lane = threadIdx.x & 31;
  const int wave = threadIdx.x >> 5;
  const int qi = blockIdx.x * 8 + wave;       // bh*2048 + s
  const int bh = qi >> 11;
  const float scale = 0.17677669529663687f;   // 32^-0.5

  const int*   ip = idx + qi * LS;
  const float* qp = q + qi * HEAD_DIM;
  const float* kb = k + (bh * S_LEN) * HEAD_DIM;
  const float* vb = v + (bh * S_LEN) * HEAD_DIM;

  // logit for neighbor l = lane & 15 (duplicated across the two halves)
  const int my_idx = ip[lane & 15];
  const float* kp = kb + my_idx * HEAD_DIM;
  float dot = 0.f;
#pragma unroll
  for (int d = 0; d < HEAD_DIM; ++d) dot += qp[d] * kp[d];
  float a = dot * scale;

  float m = a;
#pragma unroll
  for (int off = 1; off < 16; off <<= 1) m = fmaxf(m, __shfl_xor(m, off, 32));
  float e = __expf(a - m);
  float ssum = e;
#pragma unroll
  for (int off = 1; off < 16; off <<= 1) ssum += __shfl_xor(ssum, off, 32);
  const float w = e / ssum;

  float o = 0.f;
#pragma unroll
  for (int l = 0; l < LS; ++l) {
    const float wl = __shfl(w, l, 32);
    o += wl * vb[ip[l] * HEAD_DIM + lane];
  }
  out[qi * HEAD_DIM + lane] = o;
}

// ---------------------------------------------------------------------------
// Kernel 5: output projection + residual. Block = 8 m-tiles x 1 n-tile with
// the w_proj column panel staged transposed in LDS (shared by 8 waves).
// Head regather '(b h) s d -> b s (h d)' folded into A addressing.
// ---------------------------------------------------------------------------
__global__ void __launch_bounds__(256)
proj_gemm_kernel(const float* __restrict__ ao,    // [bh][s][32]
                 const float* __restrict__ w,     // [256,256]
                 const float* __restrict__ bias,  // [256]
                 const float* __restrict__ x,     // [4096,256]
                 float* __restrict__ out) {
  __shared__ float wT[16][WPAD];       // wT[col][k] = w[k][n0+col]
  const int lane = threadIdx.x & 31;
  const int wave = threadIdx.x >> 5;
  const int nt   = blockIdx.x % (D_MODEL / 16);
  const int mgrp = blockIdx.x / (D_MODEL / 16);
  const int m0 = (mgrp * 8 + wave) * 16;
  const int n0 = nt * 16;
  const int row = lane & 15;
  const int hi2 = (lane >> 4) * 2;

  {  // cooperative, coalesced panel load
    const int col = threadIdx.x & 15;
    const int kb  = threadIdx.x >> 4;
#pragma unroll
    for (int kk = 0; kk < 16; ++kk) {
      const int kidx = kb + kk * 16;
      wT[col][kidx] = w[kidx * D_MODEL + n0 + col];
    }
  }
  __syncthreads();

  const int m  = m0 + row;
  const int bb = m >> 11;
  const int s  = m & (S_LEN - 1);
  const float* abase = ao + (bb * N_HEADS * S_LEN + s) * HEAD_DIM;

  v8f acc = {};
#pragma unroll 4
  for (int k0 = 0; k0 < D_MODEL; k0 += 4) {
    const int ka = k0 + hi2;           // ka and ka+1 stay in the same head
    v2f a = *(const v2f*)(abase + (ka >> 5) * (S_LEN * HEAD_DIM) + (ka & 31));
    v2f b = *(const v2f*)&wT[row][ka];                // ds_load_b64
    acc = wmma_f32(a, b, acc);
  }

  const int n = n0 + row;
  const float bv = bias[n];
#pragma unroll
  for (int r = 0; r < 8; ++r) {
    const int mm = m0 + r + (hi2 << 2);
    out[mm * D_MODEL + n] = acc[r] + bv + x[mm * D_MODEL + n];
  }
}

// ---------------------------------------------------------------------------
extern "C" void kernel_launch(void* const* d_in, const int* in_sizes, int n_in,
                              void* d_out, int out_size, void* d_ws,
                              size_t ws_size, hipStream_t stream) {
  const float* x      = (const float*)d_in[0];
  const float* w_qkv  = (const float*)d_in[1];
  const float* b_qkv  = (const float*)d_in[2];
  const float* w_proj = (const float*)d_in[3];
  const float* b_proj = (const float*)d_in[4];
  float* out = (float*)d_out;

  // Workspace layout (~18.1 MB total, all fully written before read):
  float* q  = (float*)d_ws;                          // 16*2048*32 = 4 MB
  float* k  = q  + BH * S_LEN * HEAD_DIM;            // 4 MB
  float* v  = k  + BH * S_LEN * HEAD_DIM;            // 4 MB
  float* sq = v  + BH * S_LEN * HEAD_DIM;            // 128 KB
  float* ao = sq + BH * S_LEN;                       // 4 MB
  int* idx  = (int*)(ao + BH * S_LEN * HEAD_DIM);    // 2 MB

  qkv_gemm_kernel<<<(M_ROWS / 16 / 8) * (N3 / 16), 256, 0, stream>>>(
      x, w_qkv, b_qkv, q, k, v);
  sqnorm_kernel<<<BH * S_LEN / 256, 256, 0, stream>>>(q, sq);
  topk_kernel<<<BH * (S_LEN / 16) / 8, 256, 0, stream>>>(q, sq, idx);
  local_attn_kernel<<<BH * S_LEN / 8, 256, 0, stream>>>(q, k, v, idx, ao);
  proj_gemm_kernel<<<(M_ROWS / 16 / 8) * (D_MODEL / 16), 256, 0, stream>>>(
      ao, w_proj, b_proj, x, out);
}